// TaskAlignedLabelAssigner_12086037971198
// MI455X (gfx1250) — compile-verified
//
#include <hip/hip_runtime.h>
#include <cstdint>

#define TOPK   10
#define NCC    80      // num classes (reference NC)
#define GMAX   128     // LDS capacity for GT slots (reference G=100)
#define BLK1   256     // kernel-1 block (8 waves)
#define BLK2   128     // kernel-2 block (4 waves)  -> 45KB static LDS
#define CPITCH 84      // LDS row pitch in floats: 336B = 16B aligned, low bank conflicts

// ---------------------------------------------------------------- math helpers
__device__ __forceinline__ float clamp01f(float x) { return fminf(fmaxf(x, 0.0f), 1.0f); }
__device__ __forceinline__ float pow6f(float u)    { float u2 = u * u; return u2 * u2 * u2; }

__device__ __forceinline__ float iou_box(float px1, float py1, float px2, float py2, float parea,
                                         float gx1, float gy1, float gx2, float gy2, float garea)
{
  float ix1 = fmaxf(px1, gx1), iy1 = fmaxf(py1, gy1);
  float ix2 = fminf(px2, gx2), iy2 = fminf(py2, gy2);
  float inter = fmaxf(ix2 - ix1, 0.0f) * fmaxf(iy2 - iy1, 0.0f);
  return inter / (parea + garea - inter + 1e-7f);
}

// align = clip(cls,0,1)^0.5 * clip(iou,0,1)^6 * in_box  (identical path in both kernels)
__device__ __forceinline__ float align_metric(float cls, float iou, bool inb)
{
  float v = sqrtf(clamp01f(cls)) * pow6f(clamp01f(iou));
  return inb ? v : 0.0f;
}

// ------------------------------------------------- CDNA5 async LDS data movers
__device__ __forceinline__ void async_load_lds_b128(uint32_t lds_off, uint64_t gaddr)
{
  asm volatile("global_load_async_to_lds_b128 %0, %1, off"
               :: "v"(lds_off), "v"(gaddr) : "memory");
}
__device__ __forceinline__ void async_store_lds_b128(uint64_t gaddr, uint32_t lds_off)
{
  asm volatile("global_store_async_from_lds_b128 %0, %1, off"
               :: "v"(gaddr), "v"(lds_off) : "memory");
}
__device__ __forceinline__ void wait_async0()
{
  asm volatile("s_wait_asynccnt 0" ::: "memory");
}

// =====================================================================
// Kernel 1: for each (b,g), kth = 10th-largest align over all anchors
// =====================================================================
__global__ __launch_bounds__(BLK1)
void tal_topk_kernel(const float* __restrict__ cls_preds,
                     const float* __restrict__ box_preds,
                     const float* __restrict__ anchor_points,
                     const float* __restrict__ stride_tensor,
                     const float* __restrict__ gt_boxes,
                     const int*   __restrict__ gt_labels,
                     float* __restrict__ kth_out,
                     int A, int G)
{
  const int g   = blockIdx.x;
  const int b   = blockIdx.y;
  const int tid = threadIdx.x;

  const float4 gb  = *(const float4*)(gt_boxes + ((long)b * G + g) * 4);
  const int   lblr = gt_labels[(long)b * G + g];
  const bool  valid = lblr >= 0;
  const int   lbl   = lblr < 0 ? 0 : lblr;
  const float garea = fmaxf((gb.z - gb.x) * (gb.w - gb.y), 0.0f);

  const float*  clsb = cls_preds + ((long)b * A) * NCC + lbl;   // stride NCC column
  const float4* boxb = (const float4*)(box_preds + ((long)b * A) * 4);

  float topk[TOPK];
#pragma unroll
  for (int j = 0; j < TOPK; ++j) topk[j] = -3.402823e38f;

  for (int a = tid; a < A; a += BLK1) {
    __builtin_prefetch(clsb + (long)(a + BLK1) * NCC, 0, 0);   // global_prefetch_b8
    float2 ap = *(const float2*)(anchor_points + (long)a * 2);
    float  st = stride_tensor[a];
    float  ax = ap.x * st, ay = ap.y * st;
    bool inb = valid & (ax > gb.x) & (ax < gb.z) & (ay > gb.y) & (ay < gb.w);
    float4 pb = boxb[a];
    float  pa = fmaxf((pb.z - pb.x) * (pb.w - pb.y), 0.0f);
    float iou = iou_box(pb.x, pb.y, pb.z, pb.w, pa, gb.x, gb.y, gb.z, gb.w, garea);
    float cls = clsb[(long)a * NCC];
    float v   = align_metric(cls, iou, inb);
    // branch-free sorted insert (descending), static indices -> stays in VGPRs
#pragma unroll
    for (int j = 0; j < TOPK; ++j) {
      float mx = fmaxf(v, topk[j]);
      v        = fminf(v, topk[j]);
      topk[j]  = mx;
    }
  }

  __shared__ float s_top[BLK1 * TOPK];
#pragma unroll
  for (int j = 0; j < TOPK; ++j) s_top[tid * TOPK + j] = topk[j];

  // log2(BLK1) merge tree of sorted top-10 lists
  for (int s = BLK1 / 2; s >= 1; s >>= 1) {
    __syncthreads();
    if (tid < s) {
#pragma unroll
      for (int j = 0; j < TOPK; ++j) topk[j] = s_top[tid * TOPK + j];
#pragma unroll
      for (int j = 0; j < TOPK; ++j) {
        float v = s_top[(tid + s) * TOPK + j];
#pragma unroll
        for (int j2 = 0; j2 < TOPK; ++j2) {
          float mx = fmaxf(v, topk[j2]);
          v        = fminf(v, topk[j2]);
          topk[j2] = mx;
        }
      }
#pragma unroll
      for (int j = 0; j < TOPK; ++j) s_top[tid * TOPK + j] = topk[j];
    }
  }
  __syncthreads();
  if (tid == 0) kth_out[(long)b * G + g] = s_top[TOPK - 1];
}

// =====================================================================
// Kernel 2: per-anchor assignment, soft scores, outputs
// =====================================================================
__global__ __launch_bounds__(BLK2)
void tal_assign_kernel(const float* __restrict__ cls_preds,
                       const float* __restrict__ box_preds,
                       const float* __restrict__ anchor_points,
                       const float* __restrict__ stride_tensor,
                       const float* __restrict__ gt_boxes,
                       const int*   __restrict__ gt_labels,
                       const float* __restrict__ kth_in,
                       float* __restrict__ out_boxes,
                       float* __restrict__ out_scores,
                       float* __restrict__ out_fg,
                       float* __restrict__ out_idx,
                       int A, int G)
{
  const int b    = blockIdx.y;
  const int a0   = blockIdx.x * BLK2;
  const int tid  = threadIdx.x;
  const int a    = a0 + tid;
  const int rows = min(BLK2, A - a0);

  __shared__ float s_cls[BLK2][CPITCH];
  __shared__ float s_gt[GMAX * 4];
  __shared__ int   s_lbl[GMAX];
  __shared__ float s_kth[GMAX];

  for (int i = tid; i < G * 4; i += BLK2) s_gt[i] = gt_boxes[((long)b * G) * 4 + i];
  for (int i = tid; i < G;     i += BLK2) {
    s_lbl[i] = gt_labels[(long)b * G + i];
    s_kth[i] = kth_in  [(long)b * G + i];
  }

  // async-copy this block's cls rows (rows*NCC floats, contiguous in global)
  {
    const float* src = cls_preds + ((long)b * A + a0) * NCC;
    const int chunks = rows * (NCC / 4);
    for (int i = tid; i < chunks; i += BLK2) {
      int f = i * 4;
      int r = f / NCC, c = f % NCC;              // 16B chunks never cross a row
      uint32_t lds = (uint32_t)(uintptr_t)(&s_cls[r][c]);
      uint64_t ga  = (uint64_t)(uintptr_t)(src + f);
      async_load_lds_b128(lds, ga);
    }
    wait_async0();
  }
  __syncthreads();

  const bool lane_ok = (tid < rows);

  if (lane_ok) {
    float2 ap = *(const float2*)(anchor_points + (long)a * 2);
    float  st = stride_tensor[a];
    float  ax = ap.x * st, ay = ap.y * st;
    float4 pb = *(const float4*)(box_preds + ((long)b * A + a) * 4);
    float  pa = fmaxf((pb.z - pb.x) * (pb.w - pb.y), 0.0f);

    int   count = 0, first_g = -1, best_g = 0;
    float best_v = -3.402823e38f, iou_best = 0.0f, miou_masked = 0.0f;

    for (int g = 0; g < G; ++g) {
      float gx1 = s_gt[g * 4 + 0], gy1 = s_gt[g * 4 + 1];
      float gx2 = s_gt[g * 4 + 2], gy2 = s_gt[g * 4 + 3];
      int  lr    = s_lbl[g];
      bool valid = lr >= 0;
      int  l     = lr < 0 ? 0 : lr;
      bool inb = valid & (ax > gx1) & (ax < gx2) & (ay > gy1) & (ay < gy2);
      float ga  = fmaxf((gx2 - gx1) * (gy2 - gy1), 0.0f);
      float iou = iou_box(pb.x, pb.y, pb.z, pb.w, pa, gx1, gy1, gx2, gy2, ga);
      float cls = s_cls[tid][l];                 // LDS gather, own row
      float av  = align_metric(cls, iou, inb);
      if (av > best_v) { best_v = av; best_g = g; iou_best = iou; }  // first-max argmax
      bool m = inb && (av >= s_kth[g]);
      if (m) {
        ++count;
        if (first_g < 0) first_g = g;
        miou_masked = fmaxf(miou_masked, iou);
      }
    }

    int assigned; bool fg; float miou;
    if (count > 1) { assigned = best_g; fg = true; miou = iou_best; }          // conflict -> one_hot_best
    else           { assigned = (first_g >= 0) ? first_g : 0; fg = (count > 0); miou = miou_masked; }

    float soft = (best_v / fmaxf(best_v, 1e-9f)) * miou;   // = max_align/clip(max_align,EPS)*max_iou
    float fgf  = fg ? 1.0f : 0.0f;
    float val  = soft * fgf;

    long oa = (long)b * A + a;
    float4 tb;
    tb.x = s_gt[assigned * 4 + 0] * fgf;
    tb.y = s_gt[assigned * 4 + 1] * fgf;
    tb.z = s_gt[assigned * 4 + 2] * fgf;
    tb.w = s_gt[assigned * 4 + 3] * fgf;
    ((float4*)out_boxes)[oa] = tb;
    out_fg [oa] = fgf;
    out_idx[oa] = fg ? (float)assigned : 0.0f;

    // build one-hot score row in my own LDS row (reuse cls staging)
    float4* rowv = (float4*)(&s_cls[tid][0]);
#pragma unroll
    for (int c4 = 0; c4 < NCC / 4; ++c4) rowv[c4] = make_float4(0.f, 0.f, 0.f, 0.f);
    int pl = s_lbl[assigned]; pl = pl < 0 ? 0 : pl;
    s_cls[tid][pl] = val;
  }
  __syncthreads();

  // async-store score rows LDS -> global (fully coalesced rows*NCC floats)
  {
    float* dst = out_scores + ((long)b * A + a0) * NCC;
    const int chunks = rows * (NCC / 4);
    for (int i = tid; i < chunks; i += BLK2) {
      int f = i * 4;
      int r = f / NCC, c = f % NCC;
      uint32_t lds = (uint32_t)(uintptr_t)(&s_cls[r][c]);
      uint64_t ga  = (uint64_t)(uintptr_t)(dst + f);
      async_store_lds_b128(ga, lds);
    }
    wait_async0();
  }
}

// =====================================================================
extern "C" void kernel_launch(void* const* d_in, const int* in_sizes, int n_in,
                              void* d_out, int out_size, void* d_ws, size_t ws_size,
                              hipStream_t stream)
{
  (void)n_in; (void)out_size; (void)ws_size;

  const float* cls = (const float*)d_in[0];   // (B, A, NC)
  const float* box = (const float*)d_in[1];   // (B, A, 4)
  const float* ap  = (const float*)d_in[2];   // (A, 2)
  const float* st  = (const float*)d_in[3];   // (A, 1)
  const float* gtb = (const float*)d_in[4];   // (B, G, 4)
  const int*   gtl = (const int*)  d_in[5];   // (B, G)

  const int A  = in_sizes[3];
  const int Bn = in_sizes[1] / (A * 4);
  const int G  = in_sizes[5] / Bn;

  float* kth = (float*)d_ws;                  // B*G floats (12.8 KB)

  float* out        = (float*)d_out;          // concat: boxes | scores | fg | idx
  float* out_boxes  = out;
  float* out_scores = out + (size_t)Bn * A * 4;
  float* out_fg     = out_scores + (size_t)Bn * A * NCC;
  float* out_idx    = out_fg + (size_t)Bn * A;

  dim3 grid1(G, Bn);
  tal_topk_kernel<<<grid1, dim3(BLK1), 0, stream>>>(cls, box, ap, st, gtb, gtl, kth, A, G);

  dim3 grid2((A + BLK2 - 1) / BLK2, Bn);
  tal_assign_kernel<<<grid2, dim3(BLK2), 0, stream>>>(cls, box, ap, st, gtb, gtl, kth,
                                                      out_boxes, out_scores, out_fg, out_idx,
                                                      A, G);
}